// LowpassDetector_40295383171735
// MI455X (gfx1250) — compile-verified
//
#include <hip/hip_runtime.h>

// ---------------------------------------------------------------------------
// Compile-time Butterworth(4, Wn=0.25) design in double precision, mirroring
// the reference butter_lowpass(), then state-space blocking matrices:
//   z' = A z + B x,  y = C z + D x   (direct-form II transposed state)
//   Per 16-step tile:  y = Hx(16x16) * X  +  Hz(16x4) * z0
//                      z16 = T16(4x4) * z0 + S(4x16) * X
// All constants are emitted in WMMA fragment layout (ISA 7.12.2), and the
// state-update coefficients are pre-permuted per lane so the kernel's inner
// loops use only compile-time register indices (no v_cndmask select trees).
// ---------------------------------------------------------------------------

namespace cexpr {

constexpr double csqrt(double x) {
  double g = x > 1.0 ? x : 1.0;
  for (int i = 0; i < 64; ++i) g = 0.5 * (g + x / g);
  return g;
}

struct Cplx { double re, im; };
constexpr Cplx cadd(Cplx a, Cplx b) { return {a.re + b.re, a.im + b.im}; }
constexpr Cplx csub(Cplx a, Cplx b) { return {a.re - b.re, a.im - b.im}; }
constexpr Cplx cdiv(Cplx a, Cplx b) {
  double d = b.re * b.re + b.im * b.im;
  return {(a.re * b.re + a.im * b.im) / d, (a.im * b.re - a.re * b.im) / d};
}

struct Coeffs { double b[5]; double a[5]; };

constexpr Coeffs make_coeffs() {
  const double SQRT2 = csqrt(2.0);
  const double c8 = csqrt((1.0 + SQRT2 * 0.5) * 0.5);  // cos(pi/8)
  const double s8 = csqrt((1.0 - SQRT2 * 0.5) * 0.5);  // sin(pi/8)
  const double warped = 4.0 * (SQRT2 - 1.0);           // 2*fs*tan(pi*Wn/fs), fs=2
  Cplx p1{-s8 * warped, c8 * warped};   // m = +/-3 pair (upper member)
  Cplx p2{-c8 * warped, s8 * warped};   // m = +/-1 pair
  const double k = warped * warped * warped * warped;
  const Cplx four{4.0, 0.0};
  Cplx pz1 = cdiv(cadd(four, p1), csub(four, p1));
  Cplx pz2 = cdiv(cadd(four, p2), csub(four, p2));
  const double s1 = 2.0 * pz1.re, q1 = pz1.re * pz1.re + pz1.im * pz1.im;
  const double s2 = 2.0 * pz2.re, q2 = pz2.re * pz2.re + pz2.im * pz2.im;
  Coeffs c{};
  c.a[0] = 1.0;
  c.a[1] = -(s1 + s2);
  c.a[2] = q1 + q2 + s1 * s2;
  c.a[3] = -(s1 * q2 + s2 * q1);
  c.a[4] = q1 * q2;
  const double d1 = (4.0 - p1.re) * (4.0 - p1.re) + p1.im * p1.im;
  const double d2 = (4.0 - p2.re) * (4.0 - p2.re) + p2.im * p2.im;
  const double kz = k / (d1 * d2);
  c.b[0] = kz; c.b[1] = 4.0 * kz; c.b[2] = 6.0 * kz; c.b[3] = 4.0 * kz; c.b[4] = kz;
  return c;
}

struct Frags {
  float AHx[4][2][32];   // A-fragments of 4 K-slices of Hx: [slice][vreg][lane]
  float AHz[2][32];      // A-fragment of Hz (16x4): [vreg][lane]
  float Tloc[2][2][32];  // T16[2h+a][2h+v]        : [a][v][lane]
  float Trem[2][2][32];  // T16[2h+a][2(1-h)+v]    : [a][v][lane]
  float Sloc[2][8][32];  // S[2h+a][local slot i]  : [a][i][lane]
  float Srem[2][8][32];  // S[2h+a][remote slot i] : [a][i][lane]
};

constexpr Frags make_frags() {
  const Coeffs co = make_coeffs();
  const double b0 = co.b[0];
  double A[4][4] = {};
  double Bv[4] = {};
  for (int k = 0; k < 4; ++k) {
    A[k][0] = -co.a[k + 1];
    Bv[k] = co.b[k + 1] - co.a[k + 1] * b0;
  }
  A[0][1] = 1.0; A[1][2] = 1.0; A[2][3] = 1.0;

  // g[d] = C A^d B  (C = e0)
  double g[15] = {};
  {
    double w[4] = {Bv[0], Bv[1], Bv[2], Bv[3]};
    for (int d = 0; d < 15; ++d) {
      g[d] = w[0];
      double w2[4] = {};
      for (int r = 0; r < 4; ++r)
        for (int c = 0; c < 4; ++c) w2[r] += A[r][c] * w[c];
      for (int r = 0; r < 4; ++r) w[r] = w2[r];
    }
  }
  double Hx[16][16] = {};
  for (int t = 0; t < 16; ++t)
    for (int s = 0; s <= t; ++s) Hx[t][s] = (t == s) ? b0 : g[t - s - 1];

  double Hz[16][4] = {};
  {
    double u[4] = {1.0, 0.0, 0.0, 0.0};  // row vector C
    for (int t = 0; t < 16; ++t) {
      for (int k = 0; k < 4; ++k) Hz[t][k] = u[k];
      double u2[4] = {};
      for (int c = 0; c < 4; ++c)
        for (int r = 0; r < 4; ++r) u2[c] += u[r] * A[r][c];
      for (int c = 0; c < 4; ++c) u[c] = u2[c];
    }
  }

  double Sm[4][16] = {};
  {
    double w[4] = {Bv[0], Bv[1], Bv[2], Bv[3]};
    for (int s = 15; s >= 0; --s) {
      for (int r = 0; r < 4; ++r) Sm[r][s] = w[r];
      if (s > 0) {
        double w2[4] = {};
        for (int r = 0; r < 4; ++r)
          for (int c = 0; c < 4; ++c) w2[r] += A[r][c] * w[c];
        for (int r = 0; r < 4; ++r) w[r] = w2[r];
      }
    }
  }

  double T[4][4] = {{1, 0, 0, 0}, {0, 1, 0, 0}, {0, 0, 1, 0}, {0, 0, 0, 1}};
  for (int it = 0; it < 16; ++it) {
    double T2[4][4] = {};
    for (int r = 0; r < 4; ++r)
      for (int c = 0; c < 4; ++c)
        for (int k = 0; k < 4; ++k) T2[r][c] += A[r][k] * T[k][c];
    for (int r = 0; r < 4; ++r)
      for (int c = 0; c < 4; ++c) T[r][c] = T2[r][c];
  }

  Frags f = {};
  for (int lane = 0; lane < 32; ++lane) {
    const int m = lane & 15;
    const int h = lane >> 4;
    const int kb = h * 2;
    // WMMA A-fragments (ISA 7.12.2: lane holds M = l&15, K = (l>>4)*2 + v).
    for (int s = 0; s < 4; ++s)
      for (int v = 0; v < 2; ++v)
        f.AHx[s][v][lane] = (float)Hx[m][4 * s + kb + v];
    for (int v = 0; v < 2; ++v) f.AHz[v][lane] = (float)Hz[m][kb + v];
    // State-update coefficients permuted into this lane's slot order.
    // Local x slots:  xv[i] = x(4*(i>>1) + 2h + (i&1)),  i = 0..7
    // Remote x slots: xw[i] = x(4*(i>>1) + 2(1-h) + (i&1))
    for (int a = 0; a < 2; ++a) {
      const int r = 2 * h + a;
      for (int v = 0; v < 2; ++v) {
        f.Tloc[a][v][lane] = (float)T[r][2 * h + v];
        f.Trem[a][v][lane] = (float)T[r][2 * (1 - h) + v];
      }
      for (int i = 0; i < 8; ++i) {
        f.Sloc[a][i][lane] = (float)Sm[r][4 * (i >> 1) + 2 * h + (i & 1)];
        f.Srem[a][i][lane] = (float)Sm[r][4 * (i >> 1) + 2 * (1 - h) + (i & 1)];
      }
    }
  }
  return f;
}

}  // namespace cexpr

__device__ __constant__ cexpr::Frags g_frags = cexpr::make_frags();

typedef __attribute__((ext_vector_type(2))) float v2f;
typedef __attribute__((ext_vector_type(8))) float v8f;

// Exchange value with lane^16 (group-of-32 ds_swizzle, xor_mask=0x10, and=0x1f).
__device__ __forceinline__ float swap16(float v) {
  int i = __builtin_bit_cast(int, v);
  i = __builtin_amdgcn_ds_swizzle(i, 0x401F);
  return __builtin_bit_cast(float, i);
}

#define TT 65536
#define CH 512
#define CHUNK 256  // timesteps per wave; also the warm-up halo length
#define TILE 16

__global__ __launch_bounds__(256) void butter_wmma(const float* __restrict__ sig,
                                                   float* __restrict__ out) {
  const int lane = threadIdx.x & 31;
  const int gid  = blockIdx.x * 8 + (threadIdx.x >> 5);
  const int chunk = gid >> 5;        // 0..255  (time chunk)
  const int cg    = gid & 31;        // 0..31   (channel group of 16)
  const int n     = lane & 15;       // column / channel within group
  const int half  = lane >> 4;       // 0: K rows 0,1 ; 1: K rows 2,3
  const size_t cbase = (size_t)cg * 16 + n;

  const float* __restrict__ re = sig;
  const float* __restrict__ im = sig + (size_t)TT * CH;

  // ---- hoist constants into registers (all lane-indexed, loop-invariant) ----
  v2f aHx[4];
#pragma unroll
  for (int s = 0; s < 4; ++s) {
    aHx[s].x = g_frags.AHx[s][0][lane];
    aHx[s].y = g_frags.AHx[s][1][lane];
  }
  v2f aHz;
  aHz.x = g_frags.AHz[0][lane];
  aHz.y = g_frags.AHz[1][lane];

  float tl[2][2], tr[2][2], sl[2][8], sr[2][8];
#pragma unroll
  for (int a = 0; a < 2; ++a) {
#pragma unroll
    for (int v = 0; v < 2; ++v) {
      tl[a][v] = g_frags.Tloc[a][v][lane];
      tr[a][v] = g_frags.Trem[a][v][lane];
    }
#pragma unroll
    for (int i = 0; i < 8; ++i) {
      sl[a][i] = g_frags.Sloc[a][i][lane];
      sr[a][i] = g_frags.Srem[a][i][lane];
    }
  }

  // State fragment in WMMA B-layout for the 4x16 Z matrix:
  // vreg v of lane l holds z_{(l>>4)*2+v}(channel l&15).
  v2f z = {0.0f, 0.0f};

  const int tstart = chunk * CHUNK;
  const int halo = (chunk == 0) ? 0 : CHUNK;  // |pole|^256 ~ 1e-31: exact in f32

  // ---- warm-up halo: state updates only ----
#pragma unroll 1
  for (int tb = tstart - halo; tb < tstart; tb += TILE) {
    float xv[8];
#pragma unroll
    for (int i = 0; i < 8; ++i) {
      const size_t off = (size_t)(tb + 4 * (i >> 1) + 2 * half + (i & 1)) * CH + cbase;
      const float r = re[off], q = im[off];
      xv[i] = __builtin_fmaf(r, r, q * q);
    }
    float xw[8];
#pragma unroll
    for (int i = 0; i < 8; ++i) xw[i] = swap16(xv[i]);
    const float zwx = swap16(z.x), zwy = swap16(z.y);
    float nz0 = tl[0][0] * z.x, nz1 = tl[1][0] * z.x;
    nz0 = __builtin_fmaf(tl[0][1], z.y, nz0);  nz1 = __builtin_fmaf(tl[1][1], z.y, nz1);
    nz0 = __builtin_fmaf(tr[0][0], zwx, nz0);  nz1 = __builtin_fmaf(tr[1][0], zwx, nz1);
    nz0 = __builtin_fmaf(tr[0][1], zwy, nz0);  nz1 = __builtin_fmaf(tr[1][1], zwy, nz1);
#pragma unroll
    for (int i = 0; i < 8; ++i) {
      nz0 = __builtin_fmaf(sl[0][i], xv[i], nz0);
      nz1 = __builtin_fmaf(sl[1][i], xv[i], nz1);
      nz0 = __builtin_fmaf(sr[0][i], xw[i], nz0);
      nz1 = __builtin_fmaf(sr[1][i], xw[i], nz1);
    }
    z.x = nz0; z.y = nz1;
  }

  // ---- main chunk: WMMA output tiles + state propagation ----
#pragma unroll 1
  for (int tb = tstart; tb < tstart + CHUNK; tb += TILE) {
    if (tb + TILE < TT) {
      __builtin_prefetch(re + (size_t)(tb + TILE + 2 * half) * CH + cbase, 0, 0);
      __builtin_prefetch(im + (size_t)(tb + TILE + 2 * half) * CH + cbase, 0, 0);
    }
    float xv[8];
#pragma unroll
    for (int i = 0; i < 8; ++i) {
      const size_t off = (size_t)(tb + 4 * (i >> 1) + 2 * half + (i & 1)) * CH + cbase;
      const float r = re[off], q = im[off];
      xv[i] = __builtin_fmaf(r, r, q * q);
    }

    // y(16x16) = Hz(16x4) * z(4x16) + sum_s Hx_s(16x4) * X_s(4x16)
    v8f acc = {};
    acc = __builtin_amdgcn_wmma_f32_16x16x4_f32(false, aHz, false, z,
                                                (short)0, acc, false, false);
#pragma unroll
    for (int s = 0; s < 4; ++s) {
      v2f bx;
      bx.x = xv[2 * s];
      bx.y = xv[2 * s + 1];
      acc = __builtin_amdgcn_wmma_f32_16x16x4_f32(false, aHx[s], false, bx,
                                                  (short)0, acc, false, false);
    }
    // D layout: vreg d -> time row d + 8*half, column n.
#pragma unroll
    for (int d = 0; d < 8; ++d)
      out[(size_t)(tb + d + 8 * half) * CH + cbase] = acc[d];

    // z <- T16 z + S x   (all register indices compile-time)
    float xw[8];
#pragma unroll
    for (int i = 0; i < 8; ++i) xw[i] = swap16(xv[i]);
    const float zwx = swap16(z.x), zwy = swap16(z.y);
    float nz0 = tl[0][0] * z.x, nz1 = tl[1][0] * z.x;
    nz0 = __builtin_fmaf(tl[0][1], z.y, nz0);  nz1 = __builtin_fmaf(tl[1][1], z.y, nz1);
    nz0 = __builtin_fmaf(tr[0][0], zwx, nz0);  nz1 = __builtin_fmaf(tr[1][0], zwx, nz1);
    nz0 = __builtin_fmaf(tr[0][1], zwy, nz0);  nz1 = __builtin_fmaf(tr[1][1], zwy, nz1);
#pragma unroll
    for (int i = 0; i < 8; ++i) {
      nz0 = __builtin_fmaf(sl[0][i], xv[i], nz0);
      nz1 = __builtin_fmaf(sl[1][i], xv[i], nz1);
      nz0 = __builtin_fmaf(sr[0][i], xw[i], nz0);
      nz1 = __builtin_fmaf(sr[1][i], xw[i], nz1);
    }
    z.x = nz0; z.y = nz1;
  }
}

extern "C" void kernel_launch(void* const* d_in, const int* in_sizes, int n_in,
                              void* d_out, int out_size, void* d_ws, size_t ws_size,
                              hipStream_t stream) {
  (void)in_sizes; (void)n_in; (void)out_size; (void)d_ws; (void)ws_size;
  const float* sig = (const float*)d_in[0];
  float* out = (float*)d_out;
  // 256 chunks * 32 channel-groups = 8192 waves; 8 waves (256 thr) per block.
  dim3 grid(1024), block(256);
  hipLaunchKernelGGL(butter_wmma, grid, block, 0, stream, sig, out);
}